// FastUserEmbedding_34763465293992
// MI455X (gfx1250) — compile-verified
//
#include <hip/hip_runtime.h>
#include <hip/hip_bf16.h>
#include <math.h>

typedef __attribute__((ext_vector_type(2))) float v2f;
typedef __attribute__((ext_vector_type(8))) float v8f;

#define SEQ_L 200
#define DIM   128

// One workgroup (256 threads = 8 waves of 32) per batch row.
// LDS layout (floats): emb[25600] | w[128] | attn[208] | pool[128] | red[16]
__global__ __launch_bounds__(256) void attnpool_ln_kernel(
    const float* __restrict__ emb, const int* __restrict__ lengths,
    const float* __restrict__ w_att, const float* __restrict__ b_att,
    const float* __restrict__ gamma, const float* __restrict__ beta,
    float* __restrict__ out)
{
    extern __shared__ float smem[];
    float* s_emb  = smem;                 // 25600 floats (100 KB tile)
    float* s_w    = smem + SEQ_L * DIM;   // 128
    float* s_attn = s_w + DIM;            // 208 (padded; [200..207] zeroed)
    float* s_pool = s_attn + 208;         // 128
    float* s_red  = s_pool + DIM;         // 16

    const int b    = blockIdx.x;
    const int tid  = threadIdx.x;
    const int lane = tid & 31;
    const int wave = tid >> 5;

    // ---- Phase 0: single global read of the 200x128 tile into LDS ----
    {
        const float4* gsrc = (const float4*)(emb + (size_t)b * (SEQ_L * DIM));
        float4* ldst = (float4*)s_emb;
        #pragma unroll 5
        for (int i = tid; i < (SEQ_L * DIM) / 4; i += 256) ldst[i] = gsrc[i];
        if (tid < DIM / 4) ((float4*)s_w)[tid] = ((const float4*)w_att)[tid];
        if (tid >= SEQ_L && tid < 208) s_attn[tid] = 0.f;   // zero the pad
    }
    __syncthreads();

    // ---- Phase 1: masked attention scores (one thread per l) ----
    const int   len  = lengths[b];
    const float bias = b_att[0];
    if (tid < SEQ_L) {
        float acc = 0.f;
        const int base = tid * DIM;
        const int rot  = (tid & 31) * 4;   // rotate start to dodge bank conflicts
        #pragma unroll 8
        for (int i = 0; i < DIM / 4; ++i) {
            const int d = (rot + i * 4) & (DIM - 1);
            const float4 e  = *(const float4*)(s_emb + base + d);
            const float4 wv = *(const float4*)(s_w + d);
            acc += e.x * wv.x + e.y * wv.y + e.z * wv.z + e.w * wv.w;
        }
        s_attn[tid] = (tid < len) ? (acc + bias) : -INFINITY;
    }
    __syncthreads();

    // ---- Phase 2: softmax over L, done by wave 0 (wave32 shuffles) ----
    if (wave == 0) {
        float m = -INFINITY;
        for (int l = lane; l < SEQ_L; l += 32) m = fmaxf(m, s_attn[l]);
        for (int off = 16; off > 0; off >>= 1) m = fmaxf(m, __shfl_xor(m, off, 32));
        float s = 0.f;
        for (int l = lane; l < SEQ_L; l += 32) {
            const float e = __expf(s_attn[l] - m);   // exp(-inf)=0 for masked l
            s_attn[l] = e;
            s += e;
        }
        for (int off = 16; off > 0; off >>= 1) s += __shfl_xor(s, off, 32);
        const float inv = 1.f / s;
        for (int l = lane; l < SEQ_L; l += 32) s_attn[l] *= inv;
    }
    __syncthreads();

    // ---- Phase 3: pooled = attn(1xL) x emb(LxD) via v_wmma_f32_16x16x4_f32 ----
    // Each wave owns a 16-wide slice of D; K-loop: 50 steps of 4 (4*50 == 200).
    // A: only row M=0 carries attn -> lanes 0 (K0,K1) and 16 (K2,K3); all lanes
    //    load from the padded s_attn then mask with a select (no EXEC branching).
    // B: lane holds emb[k0 + 2*(lane>=16) + v][d0 + (lane&15)] for v=0,1.
    // D: row 0 lands in c[0], lanes 0..15.
    {
        const int  d0    = wave * 16;
        const int  col   = d0 + (lane & 15);
        const int  khalf = (lane >> 4) * 2;        // 0 for lanes 0-15, 2 for 16-31
        const int  aoff  = lane >> 3;              // lane0 -> 0, lane16 -> 2
        const bool acar  = ((lane & 15) == 0);     // lanes 0,16 carry A row 0

        v8f c = {0.f, 0.f, 0.f, 0.f, 0.f, 0.f, 0.f, 0.f};

        // prologue: load step k0 = 0
        v2f a, bv;
        {
            const float a0 = s_attn[aoff];
            const float a1 = s_attn[aoff + 1];
            a.x = acar ? a0 : 0.f;
            a.y = acar ? a1 : 0.f;
            bv.x = s_emb[khalf * DIM + col];
            bv.y = s_emb[(khalf + 1) * DIM + col];
        }

        #pragma unroll 5
        for (int k0 = 0; k0 < SEQ_L; k0 += 4) {
            // prefetch step k0+4 (clamped to 0 on the last trip; values unused)
            const int kn = (k0 + 4 < SEQ_L) ? (k0 + 4) : 0;
            const int kk = kn + aoff;
            const float a0 = s_attn[kk];
            const float a1 = s_attn[kk + 1];
            const int  row = kn + khalf;
            v2f an, bn;
            an.x = acar ? a0 : 0.f;
            an.y = acar ? a1 : 0.f;
            bn.x = s_emb[row * DIM + col];
            bn.y = s_emb[(row + 1) * DIM + col];

            c = __builtin_amdgcn_wmma_f32_16x16x4_f32(
                    /*neg_a=*/false, a, /*neg_b=*/false, bv,
                    /*c_mod=*/(short)0, c, /*reuse_a=*/false, /*reuse_b=*/false);
            a = an; bv = bn;
        }
        if (lane < 16) s_pool[d0 + lane] = c[0];
    }
    __syncthreads();

    // ---- Phase 4: layernorm over D=128 ----
    float x = 0.f;
    if (tid < DIM) x = s_pool[tid];
    float sx = x, sxx = x * x;
    for (int off = 16; off > 0; off >>= 1) {
        sx  += __shfl_xor(sx,  off, 32);
        sxx += __shfl_xor(sxx, off, 32);
    }
    if (tid < DIM && lane == 0) { s_red[wave * 2] = sx; s_red[wave * 2 + 1] = sxx; }
    __syncthreads();
    if (tid < DIM) {
        const float tsx  = s_red[0] + s_red[2] + s_red[4] + s_red[6];
        const float tsxx = s_red[1] + s_red[3] + s_red[5] + s_red[7];
        const float mu   = tsx * (1.f / DIM);
        const float var  = tsxx * (1.f / DIM) - mu * mu;
        const float inv  = rsqrtf(var + 1e-5f);
        out[(size_t)b * DIM + tid] = (x - mu) * inv * gamma[tid] + beta[tid];
    }
}

extern "C" void kernel_launch(void* const* d_in, const int* in_sizes, int n_in,
                              void* d_out, int out_size, void* d_ws, size_t ws_size,
                              hipStream_t stream) {
    const float* emb     = (const float*)d_in[0];   // [4096, 200, 128] f32
    const int*   lengths = (const int*)d_in[1];     // [4096] int
    const float* w_att   = (const float*)d_in[2];   // [128]
    const float* b_att   = (const float*)d_in[3];   // [1]
    const float* gamma   = (const float*)d_in[4];   // [128]
    const float* beta    = (const float*)d_in[5];   // [128]
    float* out = (float*)d_out;                     // [4096, 128] f32

    const size_t shmem = (size_t)(SEQ_L * DIM + DIM + 208 + DIM + 16) * sizeof(float);
    attnpool_ln_kernel<<<dim3(4096), dim3(256), shmem, stream>>>(
        emb, lengths, w_att, b_att, gamma, beta, out);
}